// MultiheadAttentionOperator_59047210385411
// MI455X (gfx1250) — compile-verified
//
#include <hip/hip_runtime.h>
#include <stdint.h>

// ---------------- types ----------------
typedef float v2f __attribute__((ext_vector_type(2)));
typedef float v8f __attribute__((ext_vector_type(8)));
typedef unsigned int v4u __attribute__((ext_vector_type(4)));
typedef int v4i __attribute__((ext_vector_type(4)));
typedef int v8i __attribute__((ext_vector_type(8)));

// ---------------- geometry ----------------
#define BTN   128        // B*T
#define NMODE 288        // 24 kx * 12 ky
#define MROW  576        // NMODE * 2 (re/im)
#define KHW   2048       // 32*64 spatial points per half-field
#define FROW  36864      // MROW*64 floats: row stride of F / Qhat / Khat / Vhat
// weight tensor (2,2,64,16,12,12,4) strides in floats
#define WS_PART 1179648
#define WS_R    589824
#define WS_I    9216
#define WS_O    576
#define WS_KX   48
#define WS_KY   4
// Parseval coefficients (irfft2 drops imag at ky==0 -> pair-mixed term)
#define C_KY1 2.384185791015625e-07f   // 1/2048^2
#define C_KY0 5.9604644775390625e-08f  // 1/(2048^2 * 4)

// ---------------- WMMA helper (fp32 16x16x4) ----------------
__device__ __forceinline__ v8f wmma4(v2f a, v2f b, v8f c) {
  return __builtin_amdgcn_wmma_f32_16x16x4_f32(false, a, false, b, (short)0, c,
                                               false, false);
}
__device__ __forceinline__ v8f zero8() {
  v8f z = {0.f, 0.f, 0.f, 0.f, 0.f, 0.f, 0.f, 0.f};
  return z;
}
// A fragment 16x4 from LDS row-major [row][k], rowstride rs
__device__ __forceinline__ v2f frag_a(const float* sA, int rs, int m0, int k0) {
  int lane = threadIdx.x & 31;
  const float* p = sA + (m0 + (lane & 15)) * rs + k0 + ((lane >> 4) << 1);
  v2f a; a.x = p[0]; a.y = p[1];
  return a;
}
// B fragment 4x16 from LDS row-major [k][n], rowstride rs
__device__ __forceinline__ v2f frag_b(const float* sB, int rs, int k0, int n0) {
  int lane = threadIdx.x & 31;
  int col = n0 + (lane & 15);
  int kb = k0 + ((lane >> 4) << 1);
  v2f b; b.x = sB[kb * rs + col]; b.y = sB[(kb + 1) * rs + col];
  return b;
}

// ---------------- Tensor Data Mover: pack D# and issue ----------------
// dims/strides in 4-byte elements (data_size code 2). Up to 4D tiles.
__device__ __forceinline__ void tdm_load_to_lds(
    const void* ldsPtr, const void* gp,
    unsigned t0, unsigned t1, unsigned t2, unsigned t3,
    unsigned d0, unsigned d1, unsigned d2, unsigned d3,
    unsigned long long s0, unsigned long long s1, unsigned long long s2,
    unsigned padEn, unsigned padInt, unsigned padAmt) {
  unsigned long long ga = (unsigned long long)(uintptr_t)gp;
  unsigned ldsOff = (unsigned)(uintptr_t)ldsPtr;  // low 32 bits = LDS offset
  v4u g0;
  g0[0] = 1u;                                         // count=1, normal mode
  g0[1] = ldsOff;                                     // lds_addr (bytes)
  g0[2] = (unsigned)ga;                               // global addr lo
  g0[3] = (unsigned)((ga >> 32) & 0x1FFFFFFu) | (2u << 30);  // hi | type=2
  v8i g1;
  g1[0] = (int)((2u << 16) | (padEn << 20) | (padInt << 22) | (padAmt << 25));
  g1[1] = (int)((d0 & 0xFFFFu) << 16);
  g1[2] = (int)((d0 >> 16) | ((d1 & 0xFFFFu) << 16));
  g1[3] = (int)((d1 >> 16) | (t0 << 16));
  g1[4] = (int)(t1 | (t2 << 16));
  g1[5] = (int)(s0 & 0xFFFFFFFFull);
  g1[6] = (int)(((s0 >> 32) & 0xFFFFull) | ((s1 & 0xFFFFull) << 16));
  g1[7] = (int)((s1 >> 16) & 0xFFFFFFFFull);
  v4i g2;
  g2[0] = (int)d2;
  g2[1] = (int)d3;
  g2[2] = (int)(s2 & 0xFFFFFFFFull);
  g2[3] = (int)(((s2 >> 32) & 0xFFFFull) | (t3 << 16));
  v4i g3; g3[0] = 0; g3[1] = 0; g3[2] = 0; g3[3] = 0;
#if __clang_major__ >= 23
  v8i g4; g4[0]=0; g4[1]=0; g4[2]=0; g4[3]=0; g4[4]=0; g4[5]=0; g4[6]=0; g4[7]=0;
  __builtin_amdgcn_tensor_load_to_lds(g0, g1, g2, g3, g4, 0);
#else
  __builtin_amdgcn_tensor_load_to_lds(g0, g1, g2, g3, 0);
#endif
}

// ================= K0a: forward DFT basis E_fwd[576][2048] =================
__global__ void k_basis_fwd(float* __restrict__ Ef) {
  int idx = blockIdx.x * 256 + threadIdx.x;
  if (idx >= MROW * KHW) return;
  int m = idx >> 11, hw = idx & 2047;
  int r = m & 1, mi = m >> 1;
  int ky = mi % 12, kxl = (mi / 12) % 12, part = mi / 144;
  int kx = part ? (20 + kxl) : kxl;
  int h = hw >> 6, w = hw & 63;
  float th = 6.283185307179586f * (kx * h * (1.0f / 32.0f) + ky * w * (1.0f / 64.0f));
  float s, c; __sincosf(th, &s, &c);
  Ef[idx] = r ? -s : c;   // e^{-i theta}
}

// ============ K0b: inverse basis E_inv[2048][576] (irfft2 semantics) ========
__global__ void k_basis_inv(float* __restrict__ Ei) {
  int idx = blockIdx.x * 256 + threadIdx.x;
  if (idx >= KHW * MROW) return;
  int p = idx / MROW, m = idx % MROW;
  int r = m & 1, mi = m >> 1;
  int ky = mi % 12, kxl = (mi / 12) % 12, part = mi / 144;
  int kx = part ? (20 + kxl) : kxl;
  int h = p >> 6, w = p & 63;
  float th = 6.283185307179586f * (kx * h * (1.0f / 32.0f) + ky * w * (1.0f / 64.0f));
  float s, c; __sincosf(th, &s, &c);
  float wgt = (ky == 0 ? 1.0f : 2.0f) * (1.0f / 2048.0f);
  Ei[idx] = r ? -wgt * s : wgt * c;  // Re( S * e^{+i psi} ), imag of ky=0 dropped
}

// ================= K0c: normalized Gaussian g[4096] =================
__global__ void k_gauss(float* __restrict__ g) {
  __shared__ float red[256];
  float vals[16]; float acc = 0.f;
#pragma unroll
  for (int i = 0; i < 16; ++i) {
    int p = threadIdx.x + i * 256;
    float dh = (float)(p >> 6) - 31.5f, dw = (float)(p & 63) - 31.5f;
    float v = __expf(-(dh * dh + dw * dw) * 0.02f);  // 0.5/sigma^2, sigma=5
    vals[i] = v; acc += v;
  }
  red[threadIdx.x] = acc; __syncthreads();
  for (int s = 128; s > 0; s >>= 1) {
    if (threadIdx.x < (unsigned)s) red[threadIdx.x] += red[threadIdx.x + s];
    __syncthreads();
  }
  float inv = 1.0f / red[0];
#pragma unroll
  for (int i = 0; i < 16; ++i) g[threadIdx.x + i * 256] = vals[i] * inv;
}

// ===== K1: forward transform, F[bt][m576][c64] = E(576x2048) @ Z(2048x64) ===
__global__ void __launch_bounds__(256) k_fwd(const float* __restrict__ z,
                                             const float* __restrict__ Ef,
                                             float* __restrict__ Fx,
                                             float* __restrict__ Fy) {
  __shared__ float sA[64 * 34];   // 64 m-rows x 32 k, TDM-padded pitch 34
  __shared__ float sB[32 * 64];   // 32 k x 64 c
  int batch = blockIdx.y;               // half*128 + bt
  int half = batch >> 7, bt = batch & 127;
  int m0blk = blockIdx.x * 64;
  const float* zsl = z + (size_t)(bt * 64 + half * 32) * 4096;  // contiguous 2048x64
  float* dst = (half ? Fy : Fx) + (size_t)bt * FROW;
  int w = threadIdx.x >> 5;
  int msub = (w & 3) * 16;
  int n0 = (w >> 2) * 32;
  v8f acc0 = zero8(), acc1 = zero8();
  for (int k0 = 0; k0 < KHW; k0 += 32) {
    if (w == 0) {
      tdm_load_to_lds(sA, Ef + (size_t)m0blk * KHW + k0,
                      32, 64, 0, 0, 32, 64, 0, 0, KHW, 0, 0, 1, 4, 1);
      tdm_load_to_lds(sB, zsl + (size_t)k0 * 64,
                      64, 32, 0, 0, 64, 32, 0, 0, 64, 0, 0, 0, 0, 0);
      __builtin_amdgcn_s_wait_tensorcnt(0);
    }
    __syncthreads();
#pragma unroll
    for (int kb = 0; kb < 32; kb += 4) {
      v2f a = frag_a(sA, 34, msub, kb);
      v2f b0 = frag_b(sB, 64, kb, n0);
      v2f b1 = frag_b(sB, 64, kb, n0 + 16);
      acc0 = wmma4(a, b0, acc0);
      acc1 = wmma4(a, b1, acc1);
    }
    __syncthreads();
  }
  int lane = threadIdx.x & 31;
  int col = lane & 15, rowoff = 8 * (lane >> 4);
#pragma unroll
  for (int j = 0; j < 8; ++j) {
    int m = m0blk + msub + j + rowoff;
    dst[(size_t)m * 64 + n0 + col] = acc0[j];
    dst[(size_t)m * 64 + n0 + 16 + col] = acc1[j];
  }
}

// == K2: per-mode complex weight GEMM: hat[bt][2mi+r][oh] (oh = o*4 + head) ==
__global__ void __launch_bounds__(256) k_wapply(
    const float* __restrict__ Fx, const float* __restrict__ Fy,
    const float* __restrict__ wq, const float* __restrict__ wk,
    const float* __restrict__ wvx, const float* __restrict__ wvy,
    float* __restrict__ Qh, float* __restrict__ Kh,
    float* __restrict__ Vxh, float* __restrict__ Vyh) {
  __shared__ float sA[128 * 130];   // 128 bt x (64 re | 64 im), pitch 130 (TDM pad)
  __shared__ float sB[2 * 64 * 64]; // [r][i=64][oh=64] via 4D TDM
  int mi = blockIdx.x;  // 0..287
  int t = blockIdx.y;   // 0:q 1:k 2:vx 3:vy
  const float* src = (t == 3) ? Fy : Fx;
  const float* wt = (t == 0) ? wq : (t == 1) ? wk : (t == 2) ? wvx : wvy;
  float* dst = (t == 0) ? Qh : (t == 1) ? Kh : (t == 2) ? Vxh : Vyh;
  int ky = mi % 12, kxl = (mi / 12) % 12, part = mi / 144;
  const float* wbase = wt + (size_t)part * WS_PART + kxl * WS_KX + ky * WS_KY;
  int w = threadIdx.x >> 5;
  if (w == 0) {
    tdm_load_to_lds(sA, src + (size_t)mi * 128,
                    128, 128, 0, 0, 128, 128, 0, 0, FROW, 0, 0, 1, 6, 1);
    // 4D tile: x=head(4,unit), y=o(16,stride 576), z=i(64,stride 9216), zz=r(2)
    tdm_load_to_lds(sB, wbase,
                    4, 16, 64, 2, 4, 16, 64, 2, WS_O, WS_I, WS_R, 0, 0, 0);
    __builtin_amdgcn_s_wait_tensorcnt(0);
  }
  __syncthreads();
  int m0 = w * 16;
  v8f cr[4], ci[4];
#pragma unroll
  for (int ns = 0; ns < 4; ++ns) { cr[ns] = zero8(); ci[ns] = zero8(); }
#pragma unroll
  for (int kb = 0; kb < 64; kb += 4) {
    v2f ar = frag_a(sA, 130, m0, kb);
    v2f ai = frag_a(sA, 130, m0, 64 + kb);
    v2f nai; nai.x = -ai.x; nai.y = -ai.y;
#pragma unroll
    for (int ns = 0; ns < 4; ++ns) {
      v2f br = frag_b(sB, 64, kb, ns * 16);
      v2f bi = frag_b(sB + 4096, 64, kb, ns * 16);
      cr[ns] = wmma4(ar, br, cr[ns]);
      cr[ns] = wmma4(nai, bi, cr[ns]);   // - Ai*Bi
      ci[ns] = wmma4(ar, bi, ci[ns]);
      ci[ns] = wmma4(ai, br, ci[ns]);
    }
  }
  int lane = threadIdx.x & 31;
  int col = lane & 15, rowoff = 8 * (lane >> 4);
  float* dre = dst + (size_t)(mi * 2) * 64;
  float* dim = dst + (size_t)(mi * 2 + 1) * 64;
#pragma unroll
  for (int ns = 0; ns < 4; ++ns)
#pragma unroll
    for (int j = 0; j < 8; ++j) {
      size_t ro = (size_t)(m0 + j + rowoff) * FROW;
      dre[ro + ns * 16 + col] = cr[ns][j];
      dim[ro + ns * 16 + col] = ci[ns][j];
    }
}

// === K3a: K~ = Parseval-weighted K (pair mixing at ky==0, coefs folded) ===
__global__ void k_ktilde(const float* __restrict__ Kh, float* __restrict__ Kt) {
  long long idx = (long long)blockIdx.x * 256 + threadIdx.x;  // < 4718592
  int m = (int)((idx >> 6) % MROW);
  int r = m & 1, mi = m >> 1;
  int ky = mi % 12, kxl = (mi / 12) % 12, part = mi / 144;
  float v = Kh[idx];
  float out;
  if (ky != 0) {
    out = C_KY1 * v;
  } else {
    int kx = part ? (20 + kxl) : kxl;
    int kx2 = (32 - kx) & 31;
    float pv = 0.f;
    if (kx2 == kx) {
      pv = v;                                  // kx==0 pairs with itself
    } else if (kx2 >= 1 && kx2 <= 11) {
      int mi2 = kx2 * 12;
      pv = Kh[idx + (long long)(mi2 - mi) * 128];
    } else if (kx2 >= 20) {
      int mi2 = 144 + (kx2 - 20) * 12;
      pv = Kh[idx + (long long)(mi2 - mi) * 128];
    }                                          // kx==20 -> no partner (12)
    out = C_KY0 * (r ? (v - pv) : (v + pv));
  }
  Kt[idx] = out;
}

// ====== K3b: scores[bn][t][s] = Q(32x9216) @ K~^T per (b,head), WMMA ======
__global__ void __launch_bounds__(128) k_scores(const float* __restrict__ Qh,
                                                const float* __restrict__ Kt,
                                                float* __restrict__ S) {
  int bn = blockIdx.x;  // b*4+n
  int b = bn >> 2, n = bn & 3;
  int w = threadIdx.x >> 5;
  int t0 = (w & 1) * 16, s0 = (w >> 1) * 16;
  int lane = threadIdx.x & 31;
  const float* Qb = Qh + (size_t)(b * 32) * FROW + n;
  const float* Kb = Kt + (size_t)(b * 32) * FROW + n;
  size_t aro = (size_t)(t0 + (lane & 15)) * FROW;
  size_t bro = (size_t)(s0 + (lane & 15)) * FROW;
  v8f acc = zero8();
  for (int k0 = 0; k0 < MROW * 16; k0 += 4) {
    int kb = k0 + ((lane >> 4) << 1);
    size_t e0 = (size_t)(kb >> 4) * 64 + (size_t)(kb & 15) * 4;  // m*64 + o*4
    v2f a; a.x = Qb[aro + e0]; a.y = Qb[aro + e0 + 4];
    v2f bb; bb.x = Kb[bro + e0]; bb.y = Kb[bro + e0 + 4];
    acc = wmma4(a, bb, acc);
  }
  float* Sb = S + (size_t)bn * 1024;
  int col = lane & 15, rowoff = 8 * (lane >> 4);
#pragma unroll
  for (int j = 0; j < 8; ++j)
    Sb[(t0 + j + rowoff) * 32 + s0 + col] = acc[j];
}

// ======= K4: mode-space mixing VM[bnt][m][o] = sum_s S * Vhat =======
__global__ void k_mix(const float* __restrict__ S, const float* __restrict__ Vh,
                      float* __restrict__ VM) {
  long long idx = (long long)blockIdx.x * 256 + threadIdx.x;  // < 4718592
  int bnt = (int)(idx / 9216);
  int rem = (int)(idx % 9216);
  int m = rem >> 4, o = rem & 15;
  int b = bnt >> 7, n = (bnt >> 5) & 3, t = bnt & 31;
  const float* Srow = S + (size_t)((b * 4 + n) * 32 + t) * 32;
  const float* vcol = Vh + (size_t)(b * 32) * FROW + (size_t)m * 64 + o * 4 + n;
  float acc = 0.f;
#pragma unroll
  for (int s = 0; s < 32; ++s) acc += Srow[s] * vcol[(size_t)s * FROW];
  VM[idx] = acc;
}

// == K5: inverse transform + Gaussian: out = g .* (E_inv(2048x576) @ VM) ==
__global__ void __launch_bounds__(256) k_inv(const float* __restrict__ Ei,
                                             const float* __restrict__ VMx,
                                             const float* __restrict__ VMy,
                                             const float* __restrict__ g,
                                             float* __restrict__ out) {
  __shared__ float sA[128 * 34];  // 128 p-rows x 32 k, TDM-padded pitch 34
  __shared__ float sB[9216];      // full VM[576][16]
  int bnt = blockIdx.y;           // ((b*4+n)*32 + t)
  int half = blockIdx.z;
  int p0 = blockIdx.x * 128;
  const float* vm = (half ? VMy : VMx) + (size_t)bnt * 9216;
  int w = threadIdx.x >> 5;
  if (w == 0) {
    tdm_load_to_lds(sB, vm, 9216, 1, 0, 0, 9216, 1, 0, 0, 9216, 0, 0, 0, 0, 0);
    __builtin_amdgcn_s_wait_tensorcnt(0);
  }
  __syncthreads();
  v8f acc = zero8();
  for (int c0 = 0; c0 < MROW; c0 += 32) {
    if (w == 0) {
      tdm_load_to_lds(sA, Ei + (size_t)p0 * MROW + c0,
                      32, 128, 0, 0, 32, 128, 0, 0, MROW, 0, 0, 1, 4, 1);
      __builtin_amdgcn_s_wait_tensorcnt(0);
    }
    __syncthreads();
    int lane = threadIdx.x & 31;
#pragma unroll
    for (int kb = 0; kb < 32; kb += 4) {
      v2f a = frag_a(sA, 34, w * 16, kb);
      int kk = c0 + kb + ((lane >> 4) << 1);
      v2f bfr;
      bfr.x = sB[kk * 16 + (lane & 15)];
      bfr.y = sB[(kk + 1) * 16 + (lane & 15)];
      acc = wmma4(a, bfr, acc);
    }
    __syncthreads();
  }
  int lane = threadIdx.x & 31;
  int o = lane & 15, rowoff = 8 * (lane >> 4);
#pragma unroll
  for (int j = 0; j < 8; ++j) {
    int prow = p0 + w * 16 + j + rowoff;
    int pg = half * 2048 + prow;
    out[((size_t)bnt * 4096 + pg) * 16 + o] = acc[j] * g[pg];
  }
}

// ================================ launch ================================
extern "C" void kernel_launch(void* const* d_in, const int* in_sizes, int n_in,
                              void* d_out, int out_size, void* d_ws,
                              size_t ws_size, hipStream_t stream) {
  (void)in_sizes; (void)n_in; (void)out_size; (void)ws_size;
  const float* z   = (const float*)d_in[0];
  const float* wq  = (const float*)d_in[1];
  const float* wk  = (const float*)d_in[2];
  const float* wvx = (const float*)d_in[3];
  const float* wvy = (const float*)d_in[4];
  float* ws = (float*)d_ws;
  float* Ef  = ws + 0;         // 1,179,648
  float* Ei  = ws + 1179648;   // 1,179,648
  float* g   = ws + 2359296;   // 4,096
  float* Fx  = ws + 2363392;   // 4,718,592
  float* Fy  = ws + 7081984;   // 4,718,592
  float* Qh  = ws + 11800576;  // 4,718,592
  float* Kh  = ws + 16519168;  // 4,718,592
  float* Vxh = ws + 21237760;  // 4,718,592
  float* Vyh = ws + 25956352;  // 4,718,592
  float* Kt  = ws + 30674944;  // 4,718,592
  float* S   = ws + 35393536;  // 16,384  (total ~141.6 MB)
  float* VMx = Qh;             // reuse after scores consumed
  float* VMy = Kh;

  k_basis_fwd<<<4608, 256, 0, stream>>>(Ef);
  k_basis_inv<<<4608, 256, 0, stream>>>(Ei);
  k_gauss<<<1, 256, 0, stream>>>(g);
  k_fwd<<<dim3(9, 256), 256, 0, stream>>>(z, Ef, Fx, Fy);
  k_wapply<<<dim3(288, 4), 256, 0, stream>>>(Fx, Fy, wq, wk, wvx, wvy,
                                             Qh, Kh, Vxh, Vyh);
  k_ktilde<<<18432, 256, 0, stream>>>(Kh, Kt);
  k_scores<<<16, 128, 0, stream>>>(Qh, Kt, S);
  k_mix<<<18432, 256, 0, stream>>>(S, Vxh, VMx);
  k_mix<<<18432, 256, 0, stream>>>(S, Vyh, VMy);
  k_inv<<<dim3(16, 512, 2), 256, 0, stream>>>(Ei, VMx, VMy, g, (float*)d_out);
}